// GCNWithAgg_14903536517580
// MI455X (gfx1250) — compile-verified
//
#include <hip/hip_runtime.h>

typedef __attribute__((ext_vector_type(2))) float v2f;
typedef __attribute__((ext_vector_type(8))) float v8f;

#define NDIM 4096
#define NNTOT (NDIM * NDIM)          // 16777216
#define R0 15099493u                 // floor(0.9*(NN-1))
#define R1 15099494u                 // ceil

__device__ __forceinline__ unsigned fkey(float x) {
  unsigned u = __float_as_uint(x);
  return (u & 0x80000000u) ? ~u : (u | 0x80000000u);
}
__device__ __forceinline__ float fdec(unsigned k) {
  unsigned u = (k & 0x80000000u) ? (k & 0x7fffffffu) : ~k;
  return __uint_as_float(u);
}
__device__ __forceinline__ unsigned umin2(unsigned a, unsigned b) { return a < b ? a : b; }

// async global -> LDS copy (CDNA5 ASYNCcnt path), inline asm for portability
__device__ __forceinline__ void async_g2l_b128(unsigned lds_off, const void* gaddr) {
  asm volatile("global_load_async_to_lds_b128 %0, %1, off"
               :: "v"(lds_off), "v"(gaddr) : "memory");
}
__device__ __forceinline__ void async_g2l_b32(unsigned lds_off, const void* gaddr) {
  asm volatile("global_load_async_to_lds_b32 %0, %1, off"
               :: "v"(lds_off), "v"(gaddr) : "memory");
}
__device__ __forceinline__ void wait_async0() {
  asm volatile("s_wait_asynccnt 0x0" ::: "memory");
}

// ---------------- init: zero histograms + select state --------------------
__global__ void k_init(unsigned* hA, unsigned* hB, unsigned* st) {
  for (int i = threadIdx.x; i < 2048; i += 256) { hA[i] = 0u; hB[i] = 0u; }
  if (threadIdx.x < 8) st[threadIdx.x] = 0u;
  if (threadIdx.x == 0) st[4] = 0xffffffffu;   // minAbove
}

// -------- agg = sum_c w[c]*attn[c] + b, fused level-0 histogram -----------
__global__ void k_agg(const float* __restrict__ attn, const float* __restrict__ w,
                      const float* __restrict__ bscal, float* __restrict__ agg,
                      unsigned* __restrict__ hist) {
  __shared__ unsigned lh[2048];
  for (int i = threadIdx.x; i < 2048; i += 256) lh[i] = 0u;
  __syncthreads();
  long long i4 = ((long long)blockIdx.x * 256 + threadIdx.x) * 4;
  float bb = bscal[0];
  float4 acc = make_float4(bb, bb, bb, bb);
#pragma unroll
  for (int c = 0; c < 12; c++) {
    const float4 v = *(const float4*)(attn + (long long)c * NNTOT + i4);
    float wc = w[c];
    acc.x += wc * v.x; acc.y += wc * v.y; acc.z += wc * v.z; acc.w += wc * v.w;
  }
  *(float4*)(agg + i4) = acc;
  atomicAdd(&lh[fkey(acc.x) >> 21], 1u);
  atomicAdd(&lh[fkey(acc.y) >> 21], 1u);
  atomicAdd(&lh[fkey(acc.z) >> 21], 1u);
  atomicAdd(&lh[fkey(acc.w) >> 21], 1u);
  __syncthreads();
  for (int i = threadIdx.x; i < 2048; i += 256) {
    unsigned v = lh[i];
    if (v) atomicAdd(&hist[i], v);
  }
}

// ---------------- radix-select passes -------------------------------------
__global__ void k_scan0(const unsigned* hist, unsigned* st) {
  if (threadIdx.x == 0) {
    unsigned cum = 0, bin = 2047, rem = 0, r = R0;
    for (int i = 0; i < 2048; i++) {
      unsigned c = hist[i];
      if (r < cum + c) { bin = (unsigned)i; rem = r - cum; break; }
      cum += c;
    }
    st[0] = bin << 21; st[1] = rem;
  }
}

__global__ void k_hist1(const float* __restrict__ agg, const unsigned* __restrict__ st,
                        unsigned* __restrict__ hist) {
  __shared__ unsigned lh[2048];
  for (int i = threadIdx.x; i < 2048; i += 256) lh[i] = 0u;
  __syncthreads();
  unsigned pfx = st[0] >> 21;
  long long i4 = ((long long)blockIdx.x * 256 + threadIdx.x) * 4;
  const float4 v = *(const float4*)(agg + i4);
  unsigned k;
  k = fkey(v.x); if ((k >> 21) == pfx) atomicAdd(&lh[(k >> 10) & 2047u], 1u);
  k = fkey(v.y); if ((k >> 21) == pfx) atomicAdd(&lh[(k >> 10) & 2047u], 1u);
  k = fkey(v.z); if ((k >> 21) == pfx) atomicAdd(&lh[(k >> 10) & 2047u], 1u);
  k = fkey(v.w); if ((k >> 21) == pfx) atomicAdd(&lh[(k >> 10) & 2047u], 1u);
  __syncthreads();
  for (int i = threadIdx.x; i < 2048; i += 256) {
    unsigned c = lh[i];
    if (c) atomicAdd(&hist[i], c);
  }
}

__global__ void k_scan1(const unsigned* hist, unsigned* st, unsigned* hA) {
  for (int i = threadIdx.x; i < 2048; i += 256) hA[i] = 0u;  // reset for pass 2
  if (threadIdx.x == 0) {
    unsigned cum = 0, bin = 2047, rem = 0, r = st[1];
    for (int i = 0; i < 2048; i++) {
      unsigned c = hist[i];
      if (r < cum + c) { bin = (unsigned)i; rem = r - cum; break; }
      cum += c;
    }
    st[0] |= bin << 10; st[1] = rem;
  }
}

__global__ void k_hist2(const float* __restrict__ agg, const unsigned* __restrict__ st,
                        unsigned* __restrict__ hist) {
  __shared__ unsigned lh[1024];
  for (int i = threadIdx.x; i < 1024; i += 256) lh[i] = 0u;
  __syncthreads();
  unsigned pfx = st[0] >> 10;
  long long i4 = ((long long)blockIdx.x * 256 + threadIdx.x) * 4;
  const float4 v = *(const float4*)(agg + i4);
  unsigned k;
  k = fkey(v.x); if ((k >> 10) == pfx) atomicAdd(&lh[k & 1023u], 1u);
  k = fkey(v.y); if ((k >> 10) == pfx) atomicAdd(&lh[k & 1023u], 1u);
  k = fkey(v.z); if ((k >> 10) == pfx) atomicAdd(&lh[k & 1023u], 1u);
  k = fkey(v.w); if ((k >> 10) == pfx) atomicAdd(&lh[k & 1023u], 1u);
  __syncthreads();
  for (int i = threadIdx.x; i < 1024; i += 256) {
    unsigned c = lh[i];
    if (c) atomicAdd(&hist[i], c);
  }
}

__global__ void k_scan2(const unsigned* hist, unsigned* st) {
  if (threadIdx.x == 0) {
    unsigned cum = 0, bin = 1023, r = st[1];
    for (int i = 0; i < 1024; i++) {
      unsigned c = hist[i];
      if (r < cum + c) { bin = (unsigned)i; break; }
      cum += c;
    }
    st[2] = st[0] | bin;  // exact key of rank R0
  }
}

// count(keys <= K0) and min(keys > K0) to get rank R1's value
__global__ void k_cnt(const float* __restrict__ agg, unsigned* __restrict__ st) {
  __shared__ unsigned sc[256];
  __shared__ unsigned sm[256];
  unsigned K0 = st[2];
  long long i4 = ((long long)blockIdx.x * 256 + threadIdx.x) * 4;
  const float4 v = *(const float4*)(agg + i4);
  unsigned cnt = 0, mn = 0xffffffffu, k;
  k = fkey(v.x); if (k <= K0) cnt++; else mn = umin2(mn, k);
  k = fkey(v.y); if (k <= K0) cnt++; else mn = umin2(mn, k);
  k = fkey(v.z); if (k <= K0) cnt++; else mn = umin2(mn, k);
  k = fkey(v.w); if (k <= K0) cnt++; else mn = umin2(mn, k);
  sc[threadIdx.x] = cnt; sm[threadIdx.x] = mn;
  __syncthreads();
  for (int o = 128; o > 0; o >>= 1) {
    if (threadIdx.x < o) {
      sc[threadIdx.x] += sc[threadIdx.x + o];
      sm[threadIdx.x] = umin2(sm[threadIdx.x], sm[threadIdx.x + o]);
    }
    __syncthreads();
  }
  if (threadIdx.x == 0) {
    atomicAdd(&st[3], sc[0]);
    atomicMin(&st[4], sm[0]);
  }
}

__global__ void k_fin(unsigned* st) {
  if (threadIdx.x == 0) {
    float v0 = fdec(st[2]);
    float v1 = (st[3] > R1) ? v0 : fdec(st[4]);
    ((float*)st)[5] = 0.5f * (v0 + v1);   // linear interp weight exactly 0.5
  }
}

// ---------------- bit-pack mask: word (u, v/32), bit v%32 -----------------
__global__ void k_pack(const float* __restrict__ agg, const unsigned* __restrict__ st,
                       unsigned* __restrict__ mw) {
  float th = ((const float*)st)[5];
  int i = blockIdx.x * 256 + threadIdx.x;
  bool p = agg[i] >= th;
  unsigned b = (unsigned)__builtin_amdgcn_ballot_w32(p);
  if ((threadIdx.x & 31) == 0) mw[i >> 5] = b;
}

// --------- deg[v] = colsum(mask | I);  s[v] = rsqrt(deg) ------------------
__global__ void k_deg(const unsigned* __restrict__ mw, float* __restrict__ sv) {
  int lane = threadIdx.x & 31;
  int g = blockIdx.x * 8 + (threadIdx.x >> 5);   // column group (32 cols)
  unsigned cnt = 0;
  for (int u = 0; u < NDIM; u++) cnt += (mw[u * 128 + g] >> lane) & 1u;
  int v = g * 32 + lane;
  unsigned db = (mw[v * 128 + g] >> lane) & 1u;  // diagonal already set?
  unsigned deg = cnt + (db ? 0u : 1u);           // self-loop fill
  sv[v] = rsqrtf((float)deg);
}

// ---------------- dense GEMM: G = (s ⊙ rows)(X @ W), f32 WMMA -------------
// one wave per 16x16 tile, K-loop in steps of 4 (V_WMMA_F32_16X16X4_F32)
__global__ void k_dense(const float* __restrict__ X, const float* __restrict__ W,
                        const float* __restrict__ sv, float* __restrict__ G,
                        int K, int Nc) {
  int lane = threadIdx.x & 31, wave = threadIdx.x >> 5;
  int half = lane >> 4, lm = lane & 15;
  int tile = blockIdx.x * (blockDim.x >> 5) + wave;
  int ntiles = Nc >> 4;
  int mt = tile / ntiles, nt = tile % ntiles;
  int row = mt * 16 + lm;
  int col = nt * 16 + lm;
  v8f acc = {};
#pragma unroll 4
  for (int k0 = 0; k0 < K; k0 += 4) {
    v2f a, b;
    a.x = X[row * K + k0 + 2 * half];
    a.y = X[row * K + k0 + 2 * half + 1];
    b.x = W[(k0 + 2 * half) * Nc + col];
    b.y = W[(k0 + 2 * half + 1) * Nc + col];
    acc = __builtin_amdgcn_wmma_f32_16x16x4_f32(false, a, false, b, (short)0, acc,
                                                false, false);
  }
#pragma unroll
  for (int j = 0; j < 8; j++) {
    int r = mt * 16 + j + 8 * half;
    G[r * Nc + col] = sv[r] * acc[j];
  }
}

// --------- mask GEMM: Out[v,c] = epi( s[v] * Σ_u A[u,v]·G[u,c] + bias[c] ) -
// block = one 128-row v-band × all Nc cols; wave w owns c-tile w and 8 v-tiles.
// B (dense G rows) and the A bitmask words are staged in LDS via CDNA5 async
// global->LDS loads (ASYNCcnt); self-loop OR'ed in branch-free so EXEC stays
// all-ones around WMMA.
__global__ void k_spgemm(const unsigned* __restrict__ mw, const float* __restrict__ G,
                         const float* __restrict__ sv, const float* __restrict__ bias,
                         float* __restrict__ Out, int Nc, int relu) {
  __shared__ float ldsB[15360];     // 60 KB
  __shared__ unsigned ldsM[512];    //  2 KB
  const int KU = 15360 / Nc;        // 60 for Nc=256, 120 for Nc=128 (mult of 4)
  int tid = threadIdx.x;
  int lane = tid & 31, wave = tid >> 5;
  int half = lane >> 4, lm = lane & 15;
  int nthreads = blockDim.x;
  int vb = blockIdx.x;              // v rows [vb*128, vb*128+128)
  int cbase = wave * 16 + lm;

  unsigned ldsB_off = (unsigned)(size_t)(void*)ldsB;   // LDS byte offset
  unsigned ldsM_off = (unsigned)(size_t)(void*)ldsM;

  v8f z = {};
  v8f acc[8];
#pragma unroll
  for (int vt = 0; vt < 8; vt++) acc[vt] = z;

  for (int u0 = 0; u0 < NDIM; u0 += KU) {
    int ku = (NDIM - u0) < KU ? (NDIM - u0) : KU;
    __syncthreads();   // previous tile's LDS reads complete workgroup-wide
    // stage dense rows G[u0 .. u0+ku) x Nc : async global->LDS, 16B/lane
    int cnt4 = (ku * Nc) >> 2;
    const float4* src = (const float4*)(G + u0 * Nc);
    for (int i = tid; i < cnt4; i += nthreads)
      async_g2l_b128(ldsB_off + (unsigned)i * 16u, (const void*)(src + i));
    // stage mask words for this v-band: 4 words per u row
    int nm = ku * 4;
    for (int i = tid; i < nm; i += nthreads)
      async_g2l_b32(ldsM_off + (unsigned)i * 4u,
                    (const void*)&mw[(u0 + (i >> 2)) * 128 + vb * 4 + (i & 3)]);
    wait_async0();     // this wave's async LDS writes are done
    __syncthreads();   // ... and everyone else's

    for (int k = 0; k < ku; k += 4) {
      v2f b;
      b.x = ldsB[(k + 2 * half) * Nc + cbase];
      b.y = ldsB[(k + 2 * half + 1) * Nc + cbase];
      int ua = u0 + k + 2 * half;
#pragma unroll
      for (int vt = 0; vt < 8; vt++) {
        unsigned w0 = ldsM[(k + 2 * half) * 4 + (vt >> 1)];
        unsigned w1 = ldsM[(k + 2 * half + 1) * 4 + (vt >> 1)];
        int sh = (vt & 1) * 16 + lm;
        int v = vb * 128 + vt * 16 + lm;
        unsigned bit0 = ((w0 >> sh) & 1u) | (unsigned)(ua == v);
        unsigned bit1 = ((w1 >> sh) & 1u) | (unsigned)(ua + 1 == v);
        v2f a;
        a.x = (float)bit0;
        a.y = (float)bit1;
        acc[vt] = __builtin_amdgcn_wmma_f32_16x16x4_f32(false, a, false, b, (short)0,
                                                        acc[vt], false, false);
      }
    }
  }

  float bc = bias[cbase];
#pragma unroll
  for (int vt = 0; vt < 8; vt++) {
#pragma unroll
    for (int j = 0; j < 8; j++) {
      int v = vb * 128 + vt * 16 + j + 8 * half;
      float val = sv[v] * acc[vt][j] + bc;
      if (relu) val = fmaxf(val, 0.0f);
      Out[v * Nc + cbase] = val;
    }
  }
}

// --------------------------------------------------------------------------
extern "C" void kernel_launch(void* const* d_in, const int* in_sizes, int n_in,
                              void* d_out, int out_size, void* d_ws, size_t ws_size,
                              hipStream_t stream) {
  const float* x     = (const float*)d_in[0];   // [4096,128]
  const float* attn  = (const float*)d_in[1];   // [1,12,4096,4096]
  const float* agg_w = (const float*)d_in[2];   // [12]
  const float* agg_b = (const float*)d_in[3];   // [1]
  const float* W1    = (const float*)d_in[4];   // [128,256]
  const float* b1    = (const float*)d_in[5];   // [256]
  const float* W2    = (const float*)d_in[6];   // [256,128]
  const float* b2    = (const float*)d_in[7];   // [128]
  float* out = (float*)d_out;                   // [4096,128]

  unsigned char* ws = (unsigned char*)d_ws;
  float*    agg   = (float*)(ws + 0);                       // 64 MB (reused below)
  unsigned* maskw = (unsigned*)(ws + 67108864);             // 2 MB
  float*    sv    = (float*)(ws + 69206016);                // 16 KB
  unsigned* hA    = (unsigned*)(ws + 69222400);             // 8 KB
  unsigned* hB    = (unsigned*)(ws + 69230592);             // 8 KB
  unsigned* st    = (unsigned*)(ws + 69238784);             // state
  // reuse of agg region after mask is packed:
  float* g = (float*)(ws + 0);                              // 4 MB max
  float* h = (float*)(ws + 4194304);                        // 4 MB

  const int blkE = NNTOT / 1024;     // 16384 blocks, 4 elems/thread
  k_init<<<1, 256, 0, stream>>>(hA, hB, st);
  k_agg<<<blkE, 256, 0, stream>>>(attn, agg_w, agg_b, agg, hA);
  k_scan0<<<1, 32, 0, stream>>>(hA, st);
  k_hist1<<<blkE, 256, 0, stream>>>(agg, st, hB);
  k_scan1<<<1, 256, 0, stream>>>(hB, st, hA);
  k_hist2<<<blkE, 256, 0, stream>>>(agg, st, hA);
  k_scan2<<<1, 32, 0, stream>>>(hA, st);
  k_cnt<<<blkE, 256, 0, stream>>>(agg, st);
  k_fin<<<1, 32, 0, stream>>>(st);
  k_pack<<<NNTOT / 256, 256, 0, stream>>>(agg, st, maskw);
  k_deg<<<16, 256, 0, stream>>>(maskw, sv);

  // layer 1: g = (s⊙)(x @ W1)  -> h = relu(s⊙(Aᵀ g) + b1)
  k_dense<<<512, 256, 0, stream>>>(x, W1, sv, g, 128, 256);      // 4096 tiles / 8 waves
  k_spgemm<<<32, 512, 0, stream>>>(maskw, g, sv, b1, h, 256, 1);
  // layer 2: g = (s⊙)(h @ W2)  -> out = s⊙(Aᵀ g) + b2
  k_dense<<<256, 256, 0, stream>>>(h, W2, sv, g, 256, 128);      // 2048 tiles / 8 waves
  k_spgemm<<<32, 256, 0, stream>>>(maskw, g, sv, b2, out, 128, 0);
}